// NewGPTEMA_90211493085427
// MI455X (gfx1250) — compile-verified
//
#include <hip/hip_runtime.h>
#include <math.h>
#include <stdint.h>

// Problem constants (from reference)
#define NDIM    8
#define EMBED   2048
#define BATCH   4
#define SEQLEN  4096

// Scan decomposition
#define SEG     32                 // segments along L
#define CLEN    (SEQLEN / SEG)     // 128 rows per segment
#define CH      16                 // rows per async chunk
#define NCHUNK  (CLEN / CH)        // 8 chunks per segment
#define TPB     256
#define WAVESZ  32
#define NWAVES  (TPB / WAVESZ)

#define SCALE_D 0.022097086912079608f   // 1/sqrt(2048)

__device__ __forceinline__ uint32_t lds_addr_of(const void* p) {
    // low 32 bits of the flat shared-aperture address are the LDS byte offset
    return (uint32_t)(uintptr_t)p;
}

// Stage one 16-row x 32-channel f32 tile (2 KB) from global into LDS using the
// CDNA5 async copy path. 4 x global_load_async_to_lds_b128: each lane moves
// 16B, 32 lanes x 4 issues = 2 KB. Tracked by ASYNCcnt (no VGPR round-trip).
__device__ __forceinline__ void async_issue_chunk(uint64_t gbase, uint32_t lbase,
                                                  int chunk, int lane) {
    const uint32_t rsub = (uint32_t)(lane >> 3);         // extra row 0..3
    const uint32_t csub = (uint32_t)(lane & 7) * 16u;    // byte offset within row
#pragma unroll
    for (int k = 0; k < 4; ++k) {
        uint64_t ga = gbase
                    + (uint64_t)((uint32_t)(chunk * CH + k * 4) + rsub) * (uint64_t)(EMBED * 4)
                    + (uint64_t)csub;
        uint32_t la = lbase + (uint32_t)(k * 512 + lane * 16);
        asm volatile("global_load_async_to_lds_b128 %0, %1, off"
                     :: "v"(la), "v"(ga) : "memory");
    }
}
__device__ __forceinline__ void wait_async_le4() {
    asm volatile("s_wait_asynccnt 0x4" ::: "memory");
}
__device__ __forceinline__ void wait_async_0() {
    asm volatile("s_wait_asynccnt 0x0" ::: "memory");
}

// ---------------------------------------------------------------------------
// Pass 1: per (b, segment, d) compute local end-states (zero initial state).
// x read here primes the 192 MB L2 for the pass-3 re-read (x = 134 MB, fits).
// ---------------------------------------------------------------------------
__global__ __launch_bounds__(TPB) void ema_pass1(const float* __restrict__ x,
                                                 const float* __restrict__ delta,
                                                 float* __restrict__ e) {
    __shared__ float sbuf[NWAVES][2][CH * WAVESZ];

    const int tid  = threadIdx.x;
    const int lane = tid & (WAVESZ - 1);
    const int wv   = tid >> 5;
    const int d    = blockIdx.x * TPB + tid;
    const int s    = blockIdx.y;
    const int b    = blockIdx.z;

    float p[NDIM], h[NDIM];
    {
        const float4* dp = (const float4*)(delta + (size_t)d * NDIM);
        float4 a = dp[0], c = dp[1];
        p[0]=a.x; p[1]=a.y; p[2]=a.z; p[3]=a.w;
        p[4]=c.x; p[5]=c.y; p[6]=c.z; p[7]=c.w;
    }
#pragma unroll
    for (int n = 0; n < NDIM; ++n) { p[n] = 1.0f / (1.0f + __expf(-p[n])); h[n] = 0.0f; }

    const int d0 = blockIdx.x * TPB + wv * WAVESZ;
    const float* gp = x + ((size_t)b * SEQLEN + (size_t)s * CLEN) * EMBED + d0;
    const uint64_t gbase = (uint64_t)(uintptr_t)gp;
    const uint32_t lb[2] = { lds_addr_of(&sbuf[wv][0][0]), lds_addr_of(&sbuf[wv][1][0]) };

    async_issue_chunk(gbase, lb[0], 0, lane);
    for (int ch = 0; ch < NCHUNK; ++ch) {
        if (ch + 1 < NCHUNK) {
            async_issue_chunk(gbase, lb[(ch + 1) & 1], ch + 1, lane);
            wait_async_le4();      // oldest (current) chunk complete
        } else {
            wait_async_0();
        }
        const float* sb = &sbuf[wv][ch & 1][0];
#pragma unroll
        for (int r = 0; r < CH; ++r) {
            float xv = sb[r * WAVESZ + lane];
#pragma unroll
            for (int n = 0; n < NDIM; ++n) h[n] = fmaf(p[n], h[n], xv);
        }
    }
    // e[((b*SEG+s)*NDIM + n)*EMBED + d]  (coalesced over d; RT -> stays in L2)
#pragma unroll
    for (int n = 0; n < NDIM; ++n)
        e[((size_t)(b * SEG + s) * NDIM + n) * EMBED + d] = h[n];
}

// ---------------------------------------------------------------------------
// Pass 2: sequential carry scan across the 32 segments, per (b,d,n).
// carry[s] = full state entering segment s;  hf' = p^CLEN * hf + e[s]
// 16 MB of L2-resident traffic; negligible.
// ---------------------------------------------------------------------------
__global__ __launch_bounds__(TPB) void ema_scan(const float* __restrict__ delta,
                                                const float* __restrict__ e,
                                                float* __restrict__ carry) {
    const int idx = blockIdx.x * TPB + threadIdx.x;   // over B*NDIM*EMBED
    const int d = idx & (EMBED - 1);
    const int n = (idx / EMBED) & (NDIM - 1);
    const int b = idx / (EMBED * NDIM);

    const float p  = 1.0f / (1.0f + __expf(-delta[(size_t)d * NDIM + n]));
    const float pC = __powf(p, (float)CLEN);

    float hf = 0.0f;
    for (int s = 0; s < SEG; ++s) {
        const size_t off = ((size_t)(b * SEG + s) * NDIM + n) * EMBED + d;
        carry[off] = hf;
        hf = fmaf(pC, hf, e[off]);
    }
}

// ---------------------------------------------------------------------------
// Pass 3: re-stream x (async->LDS, hopefully L2 hits), init states from
// carries, emit y with NON-TEMPORAL stores so the 134 MB of y does not evict
// x from the 192 MB L2 while we are still reading it.
// ---------------------------------------------------------------------------
__global__ __launch_bounds__(TPB) void ema_pass2(const float* __restrict__ x,
                                                 const float* __restrict__ delta,
                                                 const float* __restrict__ gamma,
                                                 const float* __restrict__ carry,
                                                 float* __restrict__ y) {
    __shared__ float sbuf[NWAVES][2][CH * WAVESZ];

    const int tid  = threadIdx.x;
    const int lane = tid & (WAVESZ - 1);
    const int wv   = tid >> 5;
    const int d    = blockIdx.x * TPB + tid;
    const int s    = blockIdx.y;
    const int b    = blockIdx.z;

    float p[NDIM], g[NDIM], h[NDIM];
    {
        const float4* dp = (const float4*)(delta + (size_t)d * NDIM);
        const float4* gq = (const float4*)(gamma + (size_t)d * NDIM);
        float4 a = dp[0], c = dp[1], u = gq[0], v = gq[1];
        p[0]=a.x; p[1]=a.y; p[2]=a.z; p[3]=a.w;
        p[4]=c.x; p[5]=c.y; p[6]=c.z; p[7]=c.w;
        g[0]=u.x; g[1]=u.y; g[2]=u.z; g[3]=u.w;
        g[4]=v.x; g[5]=v.y; g[6]=v.z; g[7]=v.w;
    }
#pragma unroll
    for (int n = 0; n < NDIM; ++n) {
        p[n] = 1.0f / (1.0f + __expf(-p[n]));
        g[n] *= SCALE_D;
        h[n] = carry[((size_t)(b * SEG + s) * NDIM + n) * EMBED + d];
    }

    const int d0 = blockIdx.x * TPB + wv * WAVESZ;
    const size_t rowbase = ((size_t)b * SEQLEN + (size_t)s * CLEN) * EMBED;
    const float* gp = x + rowbase + d0;
    const uint64_t gbase = (uint64_t)(uintptr_t)gp;
    const uint32_t lb[2] = { lds_addr_of(&sbuf[wv][0][0]), lds_addr_of(&sbuf[wv][1][0]) };

    async_issue_chunk(gbase, lb[0], 0, lane);
    for (int ch = 0; ch < NCHUNK; ++ch) {
        if (ch + 1 < NCHUNK) {
            async_issue_chunk(gbase, lb[(ch + 1) & 1], ch + 1, lane);
            wait_async_le4();
        } else {
            wait_async_0();
        }
        const float* sb = &sbuf[wv][ch & 1][0];
#pragma unroll
        for (int r = 0; r < CH; ++r) {
            float xv = sb[r * WAVESZ + lane];
            float acc = 0.0f;
#pragma unroll
            for (int n = 0; n < NDIM; ++n) {
                h[n] = fmaf(p[n], h[n], xv);
                acc  = fmaf(g[n], h[n], acc);
            }
            // streaming store: keep y out of L2 (x should stay resident)
            __builtin_nontemporal_store(acc,
                &y[rowbase + (size_t)(ch * CH + r) * EMBED + d]);
        }
    }
}

// ---------------------------------------------------------------------------
// Host launcher (capture-safe: kernel launches only, all on `stream`)
// ---------------------------------------------------------------------------
extern "C" void kernel_launch(void* const* d_in, const int* in_sizes, int n_in,
                              void* d_out, int out_size, void* d_ws, size_t ws_size,
                              hipStream_t stream) {
    const float* x     = (const float*)d_in[0];   // (B, L, D) f32
    const float* delta = (const float*)d_in[1];   // (D, N, 1) f32
    const float* gamma = (const float*)d_in[2];   // (D, N, 1) f32
    float* out = (float*)d_out;                   // (B, L, D) f32

    // Workspace: end-states + carries, each B*SEG*NDIM*EMBED floats (8 MB each)
    const size_t state_elems = (size_t)BATCH * SEG * NDIM * EMBED;
    float* e     = (float*)d_ws;
    float* carry = e + state_elems;

    dim3 blk(TPB);
    dim3 grid_seg(EMBED / TPB, SEG, BATCH);               // 8 x 32 x 4 blocks
    dim3 grid_scan((BATCH * NDIM * EMBED) / TPB);         // 256 blocks

    ema_pass1<<<grid_seg, blk, 0, stream>>>(x, delta, e);
    ema_scan <<<grid_scan, blk, 0, stream>>>(delta, e, carry);
    ema_pass2<<<grid_seg, blk, 0, stream>>>(x, delta, gamma, carry, out);
}